// FactorizationMachine_20040317403344
// MI455X (gfx1250) — compile-verified
//
#include <hip/hip_runtime.h>

// CDNA5 / gfx1250, wave32. FP32 WMMA path (matches f32 reference precision).

typedef __attribute__((ext_vector_type(2))) float v2f;
typedef __attribute__((ext_vector_type(8))) float v8f;

#define K_DIM     64
#define K_STEPS   16          // K_DIM / 4 (V_WMMA_F32_16X16X4_F32)
#define WAVES_PER_BLOCK 8

__global__ __launch_bounds__(256) void fm_interaction_wmma_f32(
    const float* __restrict__ U,    // (numU, 64) user embeddings
    const float* __restrict__ Mv,   // (numM, 64) movie embeddings
    float* __restrict__ out,        // (numU, numM)
    int numU, int numM)
{
    const int lane = threadIdx.x & 31;
    const int wave = threadIdx.x >> 5;
    const int row  = lane & 15;            // tile row (A: M index, B: N index)
    const int kOff = (lane >> 4) << 1;     // lanes 0-15 -> K%4 in {0,1}; 16-31 -> {2,3}

    // ---- Movie (B) tile: 16 movies x 64 dims, held in registers, reused
    // across all user tiles handled by this wave. Also its elementwise square.
    // WMMA B(4x16) layout: lane n<16 holds col N=n {K=k0,k0+1}; lane n+16 holds {k0+2,k0+3}.
    const int nBase = blockIdx.x * 16;
    const float* mrow = Mv + (size_t)(nBase + row) * K_DIM + kOff;

    v2f b[K_STEPS], b2[K_STEPS];
#pragma unroll
    for (int i = 0; i < K_STEPS; ++i) {
        b[i]  = *(const v2f*)(mrow + 4 * i);
        b2[i] = b[i] * b[i];
    }

    const int numUTiles = numU >> 4;       // 64 tiles of 16 user rows
    for (int ut = wave; ut < numUTiles; ut += WAVES_PER_BLOCK) {
        const int uBase = ut * 16;
        const float* urow = U + (size_t)(uBase + row) * K_DIM + kOff;

        // Prefetch the next user tile this wave will touch (global_prefetch_b8).
        if (ut + WAVES_PER_BLOCK < numUTiles) {
            __builtin_prefetch(urow + (size_t)WAVES_PER_BLOCK * 16 * K_DIM, 0, 1);
        }

        v8f accS = {};   // S = U * M^T
        v8f accT = {};   // T = (U*U) * (M*M)^T
#pragma unroll
        for (int i = 0; i < K_STEPS; ++i) {
            v2f a  = *(const v2f*)(urow + 4 * i);   // A(16x4) slice, K=4i..4i+3
            v2f a2 = a * a;
            // 8 args: (neg_a, A, neg_b, B, c_mod, C, reuse_a, reuse_b)
            accS = __builtin_amdgcn_wmma_f32_16x16x4_f32(
                       false, a,  false, b[i],  (short)0, accS, false, false);
            accT = __builtin_amdgcn_wmma_f32_16x16x4_f32(
                       false, a2, false, b2[i], (short)0, accT, false, false);
        }

        // Epilogue: out = 0.5 * (S^2 - T), C/D layout:
        // VGPR g -> row uBase + g + 8*(lane>=16), col nBase + (lane&15).
        const int   col   = nBase + (lane & 15);
        const int   rbase = uBase + ((lane >> 4) << 3);
        float* __restrict__ orow = out + (size_t)rbase * (size_t)numM + col;
#pragma unroll
        for (int g = 0; g < 8; ++g) {
            const float s = accS[g];
            const float t = accT[g];
            orow[(size_t)g * (size_t)numM] = 0.5f * (s * s - t);
        }
    }
}

extern "C" void kernel_launch(void* const* d_in, const int* in_sizes, int n_in,
                              void* d_out, int out_size, void* d_ws, size_t ws_size,
                              hipStream_t stream) {
    (void)n_in; (void)out_size; (void)d_ws; (void)ws_size;

    const float* U  = (const float*)d_in[0];   // (1024, 64) f32
    const float* Mv = (const float*)d_in[1];   // (50000, 64) f32
    float*       O  = (float*)d_out;           // (1024, 50000) f32

    const int numU = in_sizes[0] / K_DIM;      // 1024 rows
    const int numM = in_sizes[1] / K_DIM;      // 50000 rows

    // One block per 16-movie column tile; 8 waves per block cover the 64
    // user tiles (8 tiles each), movie tile cached in registers per wave.
    dim3 grid(numM / 16, 1, 1);                // 3125 blocks
    dim3 block(256, 1, 1);                     // 8 waves (wave32)
    fm_interaction_wmma_f32<<<grid, block, 0, stream>>>(U, Mv, O, numU, numM);
}